// Assigner_43619687858533
// MI455X (gfx1250) — compile-verified
//
#include <hip/hip_runtime.h>
#include <hip/hip_bf16.h>

typedef __attribute__((ext_vector_type(2))) float v2f;
typedef __attribute__((ext_vector_type(8))) float v8f;

#define B_    32
#define N_    32
#define A_    8400      // 6400 + 1600 + 400 anchors, = 525 * 16
#define NC_   80
#define TOPK_ 13
#define EPS_  1e-9f

// ---------------------------------------------------------------------------
// Kernel 1: bbox_scores[b,n,a] = pred_scores[b,a,lbl[b,n]] via one-hot GEMM
// using V_WMMA_F32_16X16X4_F32 (exact: one nonzero product per output).
// One wave computes one 16(n) x 16(a) tile, looping K=80 in steps of 4.
// ---------------------------------------------------------------------------
__global__ __launch_bounds__(256)
void k_bbox_scores_wmma(const float* __restrict__ ps, const int* __restrict__ glab,
                        float* __restrict__ metric) {
  const int lane = threadIdx.x & 31;
  const int wave = threadIdx.x >> 5;
  const int tile = blockIdx.x * 8 + wave;          // a-tile index
  if (tile >= A_ / 16) return;                      // wave-uniform guard (EXEC all-1s)
  const int a0 = tile * 16;
  const int n0 = blockIdx.y * 16;
  const int b  = blockIdx.z;

  // A-matrix (16x4 f32) layout: lanes 0-15 -> {K=k0,k0+1}, lanes 16-31 -> {k0+2,k0+3}
  int lab = glab[b * N_ + n0 + (lane & 15)];
  lab = lab < 0 ? 0 : (lab > NC_ - 1 ? NC_ - 1 : lab);
  const int khalf = (lane >> 4) * 2;                // 0 or 2

  // B-matrix (4x16 f32): lane holds column N=(lane&15), rows {k0+khalf, k0+khalf+1}
  const float* psb = ps + ((size_t)b * A_ + (a0 + (lane & 15))) * NC_;

  v8f acc = {};
  for (int k0 = 0; k0 < NC_; k0 += 4) {
    v2f afrag;
    afrag.x = (lab == (k0 + khalf + 0)) ? 1.0f : 0.0f;
    afrag.y = (lab == (k0 + khalf + 1)) ? 1.0f : 0.0f;
    v2f bfrag = *(const v2f*)(psb + k0 + khalf);    // 8B-aligned (even element offset)
    acc = __builtin_amdgcn_wmma_f32_16x16x4_f32(false, afrag, false, bfrag,
                                                (short)0, acc, false, false);
  }

  // C/D layout: VGPR r holds M=r (lanes 0-15) and M=r+8 (lanes 16-31), N = lane&15
  const int mhi = (lane >> 4) * 8;
  const int col = a0 + (lane & 15);
#pragma unroll
  for (int r = 0; r < 8; ++r) {
    metric[((size_t)b * N_ + (n0 + r + mhi)) * A_ + col] = acc[r];
  }
}

// ---------------------------------------------------------------------------
// Kernel 2: valid mask, pairwise IoU, align_metric = s * ov^6 (in-place).
// ---------------------------------------------------------------------------
__global__ __launch_bounds__(256)
void k_metric(const float* __restrict__ pb, const float* __restrict__ ap,
              const float* __restrict__ gb, const float* __restrict__ mgt,
              float* __restrict__ metric, float* __restrict__ ovl,
              unsigned char* __restrict__ validb) {
  size_t i = (size_t)blockIdx.x * blockDim.x + threadIdx.x;
  if (i >= (size_t)B_ * N_ * A_) return;
  int a  = (int)(i % A_);
  int bn = (int)(i / A_);
  int n  = bn % N_;
  int b  = bn / N_;

  float4 g  = ((const float4*)gb)[b * N_ + n];
  bool   mg = mgt[b * N_ + n] > 0.f;
  float2 anc = ((const float2*)ap)[a];

  float dmin = fminf(fminf(anc.x - g.x, anc.y - g.y),
                     fminf(g.z - anc.x, g.w - anc.y));
  bool valid = (dmin > EPS_) && mg;

  float4 p = ((const float4*)pb)[(size_t)b * A_ + a];
  float lx = fmaxf(g.x, p.x), ly = fmaxf(g.y, p.y);
  float rx = fminf(g.z, p.z), ry = fminf(g.w, p.w);
  float inter = fmaxf(rx - lx, 0.f) * fmaxf(ry - ly, 0.f);
  float ag  = (g.z - g.x) * (g.w - g.y);
  float apd = (p.z - p.x) * (p.w - p.y);
  float iou = inter / (ag + apd - inter + 1e-7f);

  float ov = valid ? fmaxf(iou, 0.f) : 0.f;
  float s  = valid ? metric[i] : 0.f;
  float ov2 = ov * ov;
  metric[i] = s * (ov2 * ov2 * ov2);
  ovl[i]    = ov;
  validb[i] = valid ? (unsigned char)1 : (unsigned char)0;
}

// ---------------------------------------------------------------------------
// Kernel 3: exact top-13 per (b,n) row with JAX tie-breaking (value desc,
// index asc). 13 iterative block-argmaxes; final mask = topk & valid.
// ---------------------------------------------------------------------------
__global__ __launch_bounds__(256)
void k_topk(const float* __restrict__ metric, const unsigned char* __restrict__ validb,
            unsigned char* __restrict__ maskpos) {
  const int row = blockIdx.x;                        // b*N + n
  const float* m         = metric  + (size_t)row * A_;
  const unsigned char* v = validb  + (size_t)row * A_;
  unsigned char* mp      = maskpos + (size_t)row * A_;

  __shared__ float sv[256];
  __shared__ int   si[256];
  __shared__ int   chosen[TOPK_];
  const int tid = threadIdx.x;

  for (int t = 0; t < TOPK_; ++t) {
    float bv = -1.0f;                                // metrics are >= 0
    int   bi = 0x7fffffff;
    for (int a = tid; a < A_; a += 256) {
      bool taken = false;
      for (int j = 0; j < t; ++j) taken |= (chosen[j] == a);
      if (taken) continue;
      float val = m[a];
      if (val > bv || (val == bv && a < bi)) { bv = val; bi = a; }
    }
    sv[tid] = bv; si[tid] = bi;
    __syncthreads();
    for (int s = 128; s > 0; s >>= 1) {
      if (tid < s) {
        float ov2 = sv[tid + s]; int oi = si[tid + s];
        if (ov2 > sv[tid] || (ov2 == sv[tid] && oi < si[tid])) { sv[tid] = ov2; si[tid] = oi; }
      }
      __syncthreads();
    }
    if (tid == 0) chosen[t] = si[0];
    __syncthreads();
  }

  for (int a = tid; a < A_; a += 256) {
    unsigned char sel = 0;
    for (int j = 0; j < TOPK_; ++j) sel |= (unsigned char)(chosen[j] == a);
    mp[a] = (unsigned char)(sel & v[a]);
  }
}

// ---------------------------------------------------------------------------
// Kernel 4: per (b,a) fg count, multi-GT resolution by overlaps argmax,
// rewrite mask column, emit tgi / clamped label / target_bboxes / fg.
// ---------------------------------------------------------------------------
__global__ __launch_bounds__(256)
void k_resolve(const float* __restrict__ ovl, unsigned char* __restrict__ maskpos,
               const int* __restrict__ glab, const float* __restrict__ gb,
               int* __restrict__ tgi, float* __restrict__ fg, int* __restrict__ tlab,
               float* __restrict__ out_bbox, float* __restrict__ out_fg) {
  int i = blockIdx.x * blockDim.x + threadIdx.x;
  if (i >= B_ * A_) return;
  int a = i % A_;
  int b = i / A_;

  int cnt = 0, first = 0;
  for (int n = 0; n < N_; ++n) {
    if (maskpos[((size_t)b * N_ + n) * A_ + a]) { if (cnt == 0) first = n; cnt++; }
  }

  int t; float f;
  if (cnt > 1) {
    float bv = -1.f; int bi = 0;
    for (int n = 0; n < N_; ++n) {
      float o = ovl[((size_t)b * N_ + n) * A_ + a];
      if (o > bv) { bv = o; bi = n; }                // first-max tie rule
    }
    for (int n = 0; n < N_; ++n)
      maskpos[((size_t)b * N_ + n) * A_ + a] = (unsigned char)(n == bi);
    t = bi; f = 1.f;
  } else {
    t = (cnt == 1) ? first : 0;
    f = (float)cnt;
  }

  tgi[i] = t;
  fg[i]  = f;
  int L = glab[b * N_ + t];
  L = L < 0 ? 0 : (L > NC_ - 1 ? NC_ - 1 : L);
  tlab[i] = L;
  ((float4*)out_bbox)[i] = ((const float4*)gb)[b * N_ + t];
  out_fg[i] = (f > 0.f) ? 1.f : 0.f;
}

// ---------------------------------------------------------------------------
// Kernel 5: per-(b,n) row maxes of am = metric*mask and ov*mask.
// ---------------------------------------------------------------------------
__global__ __launch_bounds__(256)
void k_rowmax(const float* __restrict__ metric, const float* __restrict__ ovl,
              const unsigned char* __restrict__ maskpos,
              float* __restrict__ pos_align, float* __restrict__ pos_ov) {
  const int row = blockIdx.x;
  const size_t base = (size_t)row * A_;
  __shared__ float s1[256], s2[256];
  const int tid = threadIdx.x;

  float m1 = 0.f, m2 = 0.f;
  for (int a = tid; a < A_; a += 256) {
    if (maskpos[base + a]) {
      m1 = fmaxf(m1, metric[base + a]);
      m2 = fmaxf(m2, ovl[base + a]);
    }
  }
  s1[tid] = m1; s2[tid] = m2;
  __syncthreads();
  for (int s = 128; s > 0; s >>= 1) {
    if (tid < s) {
      s1[tid] = fmaxf(s1[tid], s1[tid + s]);
      s2[tid] = fmaxf(s2[tid], s2[tid + s]);
    }
    __syncthreads();
  }
  if (tid == 0) { pos_align[row] = s1[0]; pos_ov[row] = s2[0]; }
}

// ---------------------------------------------------------------------------
// Kernel 6: norm[b,a] = max_n( am * pos_ov[n] / (pos_align[n] + EPS) ).
// ---------------------------------------------------------------------------
__global__ __launch_bounds__(256)
void k_norm(const float* __restrict__ metric, const unsigned char* __restrict__ maskpos,
            const float* __restrict__ pos_align, const float* __restrict__ pos_ov,
            float* __restrict__ norm) {
  int i = blockIdx.x * blockDim.x + threadIdx.x;
  if (i >= B_ * A_) return;
  int a = i % A_;
  int b = i / A_;
  float best = 0.f;                                  // all candidates >= 0
  for (int n = 0; n < N_; ++n) {
    size_t idx = ((size_t)b * N_ + n) * A_ + a;
    float am = maskpos[idx] ? metric[idx] : 0.f;
    float val = am * pos_ov[b * N_ + n] / (pos_align[b * N_ + n] + EPS_);
    best = fmaxf(best, val);
  }
  norm[i] = best;
}

// ---------------------------------------------------------------------------
// Kernel 7: target_scores = onehot(tlab) * norm where fg>0, coalesced.
// ---------------------------------------------------------------------------
__global__ __launch_bounds__(256)
void k_scores(const float* __restrict__ norm, const int* __restrict__ tlab,
              const float* __restrict__ fg, float* __restrict__ out_ts) {
  size_t i = (size_t)blockIdx.x * blockDim.x + threadIdx.x;
  if (i >= (size_t)B_ * A_ * NC_) return;
  int c  = (int)(i % NC_);
  int ba = (int)(i / NC_);
  float o = 0.f;
  if (fg[ba] > 0.f && c == tlab[ba]) o = norm[ba];
  out_ts[i] = o;
}

// ---------------------------------------------------------------------------
static inline size_t alignUp256(size_t x) { return (x + 255) & ~(size_t)255; }

extern "C" void kernel_launch(void* const* d_in, const int* in_sizes, int n_in,
                              void* d_out, int out_size, void* d_ws, size_t ws_size,
                              hipStream_t stream) {
  const float* ps   = (const float*)d_in[0];   // (B, A, NC)
  const float* pb   = (const float*)d_in[1];   // (B, A, 4)
  const float* ap   = (const float*)d_in[2];   // (A, 2)
  const int*   glab = (const int*)  d_in[3];   // (B, N, 1)
  const float* gb   = (const float*)d_in[4];   // (B, N, 4)
  const float* mgt  = (const float*)d_in[5];   // (B, N, 1)

  // workspace carve-out
  char* w = (char*)d_ws;
  size_t off = 0;
  const size_t szBNA_f = (size_t)B_ * N_ * A_ * sizeof(float);
  const size_t szBNA_b = (size_t)B_ * N_ * A_;
  const size_t szBA_f  = (size_t)B_ * A_ * sizeof(float);
  const size_t szBA_i  = (size_t)B_ * A_ * sizeof(int);

  float* metric        = (float*)(w + off); off = alignUp256(off + szBNA_f);
  float* ovl           = (float*)(w + off); off = alignUp256(off + szBNA_f);
  unsigned char* validb  = (unsigned char*)(w + off); off = alignUp256(off + szBNA_b);
  unsigned char* maskpos = (unsigned char*)(w + off); off = alignUp256(off + szBNA_b);
  int*   tgi           = (int*)  (w + off); off = alignUp256(off + szBA_i);
  float* fg            = (float*)(w + off); off = alignUp256(off + szBA_f);
  int*   tlab          = (int*)  (w + off); off = alignUp256(off + szBA_i);
  float* norm          = (float*)(w + off); off = alignUp256(off + szBA_f);
  float* pos_align     = (float*)(w + off); off = alignUp256(off + (size_t)B_ * N_ * sizeof(float));
  float* pos_ov        = (float*)(w + off); off = alignUp256(off + (size_t)B_ * N_ * sizeof(float));
  (void)ws_size; (void)in_sizes; (void)n_in; (void)out_size;

  // output carve-out: [target_bboxes | target_scores | fg]
  float* out_bbox = (float*)d_out;                               // B*A*4
  float* out_ts   = out_bbox + (size_t)B_ * A_ * 4;              // B*A*NC
  float* out_fg   = out_ts   + (size_t)B_ * A_ * NC_;            // B*A

  // 1) WMMA one-hot gather -> raw bbox_scores in `metric`
  {
    dim3 grid((A_ / 16 + 7) / 8, N_ / 16, B_);  // 66 x 2 x 32, 8 waves/block
    k_bbox_scores_wmma<<<grid, 256, 0, stream>>>(ps, glab, metric);
  }
  // 2) valid / IoU / align_metric (in place over metric)
  {
    size_t total = (size_t)B_ * N_ * A_;
    k_metric<<<(unsigned)((total + 255) / 256), 256, 0, stream>>>(
        pb, ap, gb, mgt, metric, ovl, validb);
  }
  // 3) top-13 per (b,n)
  k_topk<<<B_ * N_, 256, 0, stream>>>(metric, validb, maskpos);
  // 4) fg / tgi resolution + bbox + fg outputs
  k_resolve<<<(B_ * A_ + 255) / 256, 256, 0, stream>>>(
      ovl, maskpos, glab, gb, tgi, fg, tlab, out_bbox, out_fg);
  // 5) per-row maxes
  k_rowmax<<<B_ * N_, 256, 0, stream>>>(metric, ovl, maskpos, pos_align, pos_ov);
  // 6) norm per (b,a)
  k_norm<<<(B_ * A_ + 255) / 256, 256, 0, stream>>>(metric, maskpos, pos_align, pos_ov, norm);
  // 7) target_scores
  {
    size_t total = (size_t)B_ * A_ * NC_;
    k_scores<<<(unsigned)((total + 255) / 256), 256, 0, stream>>>(norm, tlab, fg, out_ts);
  }
}